// SSDLoss_41970420417098
// MI455X (gfx1250) — compile-verified
//
#include <hip/hip_runtime.h>

#define NB 64
#define AA 24564
#define CC 21

typedef __attribute__((ext_vector_type(2))) float v2f;
typedef __attribute__((ext_vector_type(8))) float v8f;

// ---------------------------------------------------------------------------
// CDNA5 async global->LDS copy path (guarded; fallback = b128 register loads)
// Builtin signatures (from hipcc diagnostics):
//   b128: (AS1 int_vec16*, AS3 int_vec16*, imm int, imm int)
//   b32 : (AS1 int*,       AS3 int*,       imm int, imm int)
// ---------------------------------------------------------------------------
#if __has_builtin(__builtin_amdgcn_global_load_async_to_lds_b128) && \
    __has_builtin(__builtin_amdgcn_global_load_async_to_lds_b32)
#define USE_ASYNC_LDS 1
#else
#define USE_ASYNC_LDS 0
#endif

#if USE_ASYNC_LDS
typedef int gvec4i __attribute__((vector_size(4 * sizeof(int))));
typedef __attribute__((address_space(1))) gvec4i* gbl_v4i_p;
typedef __attribute__((address_space(3))) gvec4i* lds_v4i_p;
typedef __attribute__((address_space(1))) int*    gbl_i32_p;
typedef __attribute__((address_space(3))) int*    lds_i32_p;

__device__ __forceinline__ gbl_v4i_p gbl128(const void* p) {
    return (gbl_v4i_p)(unsigned long long)p;
}
__device__ __forceinline__ lds_v4i_p lds128(void* p) {
    // low 32 bits of a generic LDS pointer are the LDS byte offset
    return (lds_v4i_p)(unsigned int)(unsigned long long)p;
}
__device__ __forceinline__ gbl_i32_p gbl32(const void* p) {
    return (gbl_i32_p)(unsigned long long)p;
}
__device__ __forceinline__ lds_i32_p lds32(void* p) {
    return (lds_i32_p)(unsigned int)(unsigned long long)p;
}
#endif

// ---------------------------------------------------------------------------
// Kernel 0: zero accumulators (d_ws is poisoned; must re-init every launch)
// ---------------------------------------------------------------------------
__global__ void ssd_init_kernel(int* __restrict__ row_numpos, float* __restrict__ accums) {
    int t = threadIdx.x;
    if (t < NB) row_numpos[t] = 0;
    if (t < 4)  accums[t] = 0.0f;   // [0]=loc_loss, [1]=pos_ce, [2]=neg_ce
}

// ---------------------------------------------------------------------------
// Kernel 1: per-anchor cross entropy. Block = 128 threads = 4 waves; each
// wave handles 32 anchors. cls logits staged to LDS at natural stride 21
// (odd => conflict-free across 64 banks), sum-of-exp done on the matrix pipe
// via V_WMMA_F32_16X16X4_F32 against an all-ones B (6 K-chunks, classes
// 21..23 masked to 0 by lane-predicated selects in the last chunk).
// ---------------------------------------------------------------------------
__global__ __launch_bounds__(128) void ssd_ce_kernel(
    const float* __restrict__ locp, const float* __restrict__ loct,
    const float* __restrict__ clsp, const int* __restrict__ tgts,
    float* __restrict__ ce_neg, int* __restrict__ row_numpos,
    float* __restrict__ accums)
{
    __shared__ __align__(16) float raw[4][32 * CC];   // 672 floats per wave
    __shared__ float mxs [4][32];
    __shared__ float sums[4][32];

    const int lane = threadIdx.x & 31;
    const int wave = threadIdx.x >> 5;
    const long long anchor0 = (long long)blockIdx.x * 128 + wave * 32;

    const float* src = clsp + anchor0 * CC;           // 672 contiguous floats
    float* myraw = &raw[wave][0];

#if USE_ASYNC_LDS
    // 5 x async b128 + 1 x async b32 per lane; tracked on ASYNCcnt.
    #pragma unroll
    for (int j = 0; j < 5; ++j) {
        const int e = (j * 32 + lane) * 4;
        __builtin_amdgcn_global_load_async_to_lds_b128(
            gbl128(src + e), lds128(myraw + e), 0, 0);
    }
    __builtin_amdgcn_global_load_async_to_lds_b32(
        gbl32(src + 640 + lane), lds32(myraw + 640 + lane), 0, 0);
#if __has_builtin(__builtin_amdgcn_s_wait_asynccnt)
    __builtin_amdgcn_s_wait_asynccnt(0);
#else
    asm volatile("s_wait_asynccnt 0" ::: "memory");
#endif
#else
    // Fallback: wide b128 loads through VGPRs, then LDS stores.
    float4 v[5];
    #pragma unroll
    for (int j = 0; j < 5; ++j) v[j] = ((const float4*)src)[j * 32 + lane];
    const float tailv = src[640 + lane];
    #pragma unroll
    for (int j = 0; j < 5; ++j)
        *(float4*)(myraw + (j * 32 + lane) * 4) = v[j];
    myraw[640 + lane] = tailv;
#endif
    __syncthreads();

    // Per-anchor max + target logit (lane <-> anchor). Stride-21 rows are
    // bank-conflict-free (gcd(21,64)==1).
    const float* arow = myraw + lane * CC;
    float mx = -1e30f;
    #pragma unroll
    for (int k = 0; k < CC; ++k) mx = fmaxf(mx, arow[k]);
    mxs[wave][lane] = mx;

    const long long anchor = anchor0 + lane;
    const int tgt = tgts[anchor];
    const int tc  = tgt < 0 ? 0 : (tgt > CC - 1 ? CC - 1 : tgt);
    const float xt = arow[tc];
    __syncthreads();

    // A layout (32-bit 16x4): lane L holds (M = L&15, K = kbase+{0,1}),
    // kbase += 2 for lanes 16..31. D: lanes 0/16 hold M=0..7 / 8..15 in v0..7.
    const v2f ones = {1.0f, 1.0f};
    const int koff = (lane >> 4) << 1;
    #pragma unroll
    for (int h = 0; h < 2; ++h) {
        const int mloc = h * 16 + (lane & 15);
        const float m  = mxs[wave][mloc];
        const float* mrow = myraw + mloc * CC;
        v8f acc = {};
        #pragma unroll
        for (int c = 0; c < 6; ++c) {
            const int k0 = 4 * c + koff;
            const int k1 = k0 + 1;
            const float x0 = mrow[k0 > CC - 1 ? CC - 1 : k0];
            const float x1 = mrow[k1 > CC - 1 ? CC - 1 : k1];
            v2f av;
            av.x = (k0 <= CC - 1) ? __expf(x0 - m) : 0.0f;
            av.y = (k1 <= CC - 1) ? __expf(x1 - m) : 0.0f;
            acc = __builtin_amdgcn_wmma_f32_16x16x4_f32(
                false, av, false, ones, (short)0, acc, false, false);
        }
        if ((lane & 15) == 0) {
            const int b = h * 16 + (lane >> 4) * 8;
            #pragma unroll
            for (int q = 0; q < 8; ++q) sums[wave][b + q] = acc[q];
        }
    }
    __syncthreads();

    const float se = sums[wave][lane];
    const float ce = __logf(se) + mx - xt;          // logsumexp - x[target]
    const bool  pos = tgt > 0;
    ce_neg[anchor] = pos ? -1.0f : ce;              // positives keyed below all CE>=0

    if (pos) {
        const int row = (int)(anchor / AA);
        atomicAdd(&row_numpos[row], 1);
        atomicAdd(&accums[1], ce);
        float s = 0.0f;
        #pragma unroll
        for (int j = 0; j < 4; ++j) {
            const float d  = locp[anchor * 4 + j] - loct[anchor * 4 + j];
            const float ad = fabsf(d);
            s += (ad < 1.0f) ? 0.5f * d * d : ad - 0.5f;
        }
        atomicAdd(&accums[0], s);
    }
}

// ---------------------------------------------------------------------------
// Kernel 2: per-row hard negative mining. Exact radix-select of the k-th
// largest CE among negatives (4x 8-bit passes, LDS histogram), then
// sum = sum(ce > T) + (k - count(ce > T)) * T.  ce_neg lives in L2 (6.3 MB).
// ---------------------------------------------------------------------------
__device__ __forceinline__ unsigned f2key(float f) {
    unsigned u = __float_as_uint(f);
    return u ^ ((u >> 31) ? 0xFFFFFFFFu : 0x80000000u);
}

__global__ __launch_bounds__(256) void ssd_topk_kernel(
    const float* __restrict__ ce_neg, const int* __restrict__ row_numpos,
    float* __restrict__ accums)
{
    __shared__ unsigned hist[256];
    __shared__ unsigned sel_b, sel_rem, gcount;
    __shared__ float    fsum;

    const int row = blockIdx.x;
    const float* base = ce_neg + (long long)row * AA;
    const int npos = row_numpos[row];
    int k = 3 * npos;
    const int negc = AA - npos;
    if (k > negc) k = negc;
    if (k <= 0) return;                      // uniform across block

    unsigned prefix = 0;
    unsigned remaining = (unsigned)k;
    for (int pass = 0; pass < 4; ++pass) {
        const int shift = 24 - 8 * pass;
        const unsigned hi_mask = (pass == 0) ? 0u : (0xFFFFFFFFu << (shift + 8));
        hist[threadIdx.x] = 0u;
        __syncthreads();
        for (int i = threadIdx.x; i < AA; i += 256) {
            const unsigned key = f2key(base[i]);
            if ((key & hi_mask) == prefix)
                atomicAdd(&hist[(key >> shift) & 0xFFu], 1u);
        }
        __syncthreads();
        if (threadIdx.x == 0) {
            unsigned c = 0, b = 0;
            for (int bb = 255; bb >= 0; --bb) {
                const unsigned c2 = c + hist[bb];
                if (c2 >= remaining) { b = (unsigned)bb; break; }
                c = c2;
            }
            sel_b = b;
            sel_rem = remaining - c;
        }
        __syncthreads();
        prefix |= sel_b << shift;
        remaining = sel_rem;
        __syncthreads();
    }

    const unsigned T = prefix;               // key of k-th largest
    if (threadIdx.x == 0) { fsum = 0.0f; gcount = 0u; }
    __syncthreads();
    float lsum = 0.0f; unsigned lcnt = 0;
    for (int i = threadIdx.x; i < AA; i += 256) {
        const float f = base[i];
        if (f2key(f) > T) { lsum += f; ++lcnt; }
    }
    atomicAdd(&fsum, lsum);
    atomicAdd(&gcount, lcnt);
    __syncthreads();
    if (threadIdx.x == 0) {
        const unsigned u = T ^ ((T & 0x80000000u) ? 0x80000000u : 0xFFFFFFFFu);
        const float vT = __uint_as_float(u);
        atomicAdd(&accums[2], fsum + (float)(k - (int)gcount) * vT);
    }
}

// ---------------------------------------------------------------------------
// Kernel 3: finalize  (loc + pos_ce + neg_ce) / num_pos
// ---------------------------------------------------------------------------
__global__ void ssd_final_kernel(const int* __restrict__ row_numpos,
                                 const float* __restrict__ accums,
                                 float* __restrict__ out)
{
    if (threadIdx.x == 0) {
        int np = 0;
        for (int r = 0; r < NB; ++r) np += row_numpos[r];
        out[0] = (accums[0] + accums[1] + accums[2]) / (float)np;
    }
}

// ---------------------------------------------------------------------------
extern "C" void kernel_launch(void* const* d_in, const int* in_sizes, int n_in,
                              void* d_out, int out_size, void* d_ws, size_t ws_size,
                              hipStream_t stream) {
    const float* locp = (const float*)d_in[0];
    const float* loct = (const float*)d_in[1];
    const float* clsp = (const float*)d_in[2];
    const int*   tgts = (const int*)  d_in[3];
    float* out = (float*)d_out;

    float* ce_neg      = (float*)d_ws;                                    // N*A floats
    int*   row_numpos  = (int*)((char*)d_ws + sizeof(float) * (size_t)NB * AA);
    float* accums      = (float*)(row_numpos + NB);                       // 4 floats

    ssd_init_kernel <<<1, 128, 0, stream>>>(row_numpos, accums);
    ssd_ce_kernel   <<<(NB * AA) / 128, 128, 0, stream>>>(
        locp, loct, clsp, tgts, ce_neg, row_numpos, accums);
    ssd_topk_kernel <<<NB, 256, 0, stream>>>(ce_neg, row_numpos, accums);
    ssd_final_kernel<<<1, 32, 0, stream>>>(row_numpos, accums, out);
}